// AdaFreqFilter_65395172049046
// MI455X (gfx1250) — compile-verified
//
#include <hip/hip_runtime.h>

// ---------------------------------------------------------------------------
// AdaFreqFilter (channel_wise): out = H - K * (H - D_in^{-1/2} A^T D_out^{-1/2} H)
//
// CSR-by-dst counting sort, then one wave per node. The per-node neighborhood
// gather (avg 12 x 512B rows, L2-resident: H = 25.6MB << 192MB L2) is done by
// the CDNA5 Tensor Data Mover in GATHER MODE: one tensor_load_to_lds per <=16
// scattered rows, indices packed 16-bit into D# groups 2/3, completion via
// s_wait_tensorcnt. Accumulation then reads LDS with ds_load_b128. No WMMA:
// the op is ~0.15 GFLOP of sparse work; matrix pipes don't apply.
// ---------------------------------------------------------------------------

__global__ void k_zero(int* __restrict__ p, int n) {
  int i = blockIdx.x * blockDim.x + threadIdx.x;
  int stride = gridDim.x * blockDim.x;
  for (; i < n; i += stride) p[i] = 0;
}

__global__ void k_deg(const int* __restrict__ src, const int* __restrict__ dst,
                      int* __restrict__ deg_out, int* __restrict__ deg_in, int e) {
  int i = blockIdx.x * blockDim.x + threadIdx.x;
  if (i < e) {
    atomicAdd(&deg_out[src[i]], 1);
    atomicAdd(&deg_in[dst[i]], 1);
  }
}

__global__ __launch_bounds__(1024) void k_scan(const int* __restrict__ deg_in,
                                               int* __restrict__ offsets, int n) {
  __shared__ int smem[1024];
  __shared__ int s_carry;
  const int tid = threadIdx.x;
  if (tid == 0) s_carry = 0;
  __syncthreads();
  for (int base = 0; base < n; base += 1024) {
    int v = (base + tid < n) ? deg_in[base + tid] : 0;
    smem[tid] = v;
    __syncthreads();
    for (int off = 1; off < 1024; off <<= 1) {
      int t = (tid >= off) ? smem[tid - off] : 0;
      __syncthreads();
      smem[tid] += t;
      __syncthreads();
    }
    const int incl = smem[tid];
    const int carry = s_carry;
    if (base + tid < n) offsets[base + tid] = carry + incl - v;  // exclusive
    __syncthreads();
    if (tid == 1023) s_carry = carry + incl;
    __syncthreads();
  }
  if (tid == 0) offsets[n] = s_carry;
}

__global__ void k_bucket(const int* __restrict__ src, const int* __restrict__ dst,
                         const int* __restrict__ offsets, int* __restrict__ cursor,
                         int* __restrict__ csr_src, int e) {
  int i = blockIdx.x * blockDim.x + threadIdx.x;
  if (i < e) {
    const int d = dst[i];
    const int slot = atomicAdd(&cursor[d], 1);
    csr_src[offsets[d] + slot] = src[i];
  }
}

#define WPB 4            // waves (nodes) per block
#define GROWS 16         // rows per TDM gather descriptor (16-bit index mode)

typedef unsigned int v4u __attribute__((ext_vector_type(4)));
typedef int v4i __attribute__((ext_vector_type(4)));
typedef int v8i __attribute__((ext_vector_type(8)));

#if __has_builtin(__builtin_amdgcn_tensor_load_to_lds) && \
    __has_builtin(__builtin_amdgcn_s_wait_tensorcnt)
#define HAVE_TDM 1
#else
#define HAVE_TDM 0
#endif

// One wave per node; lane L owns channels [4L,4L+4).
__global__ __launch_bounds__(WPB * 32) void k_agg(
    const float* __restrict__ H, const float* __restrict__ K,
    const int* __restrict__ offsets, const int* __restrict__ deg_out,
    const int* __restrict__ csr_src, float* __restrict__ out, int n) {
  __shared__ __align__(1024) float smem[WPB * GROWS * 128];  // 32KB: 8KB stage/wave

  const int lane = threadIdx.x & 31;
  const int wave = threadIdx.x >> 5;
  const int node = blockIdx.x * WPB + wave;
  if (node >= n) return;

  const float4 h  = ((const float4*)(H + (size_t)node * 128))[lane];
  const float4 kc = ((const float4*)K)[lane];

  const int base = offsets[node];
  const int cnt  = offsets[node + 1] - base;  // == deg_in[node]

  float* stage = &smem[wave * GROWS * 128];
  const unsigned lds_base =
      (unsigned)__builtin_amdgcn_readfirstlane((int)(size_t)stage);  // LDS byte offset
  const unsigned long long ga = (unsigned long long)(size_t)H;

  float4 acc = make_float4(0.f, 0.f, 0.f, 0.f);

  for (int c = 0; c < cnt; c += GROWS) {
    const int m = (cnt - c < GROWS) ? (cnt - c) : GROWS;
    // Cooperative load of up to 16 edge sources; wave-uniform via v_readlane.
    int sl = (lane < m) ? csr_src[base + c + lane] : 0;

#if HAVE_TDM
    // Pack 16-bit row indices, 2 per dword (D# groups 2/3, gather mode).
    unsigned idx[8];
#pragma unroll
    for (int j = 0; j < 8; ++j) {
      const int a = (2 * j < m) ? __builtin_amdgcn_readlane(sl, 2 * j) : 0;
      const int b = (2 * j + 1 < m) ? __builtin_amdgcn_readlane(sl, 2 * j + 1) : 0;
      idx[j] = ((unsigned)a & 0xFFFFu) | (((unsigned)b & 0xFFFFu) << 16);
    }
    // D# group 0: count=1, gather_mode=1 (bit31), 16-bit indices (bit30=0),
    //             lds_addr, global_addr[56:0], type=2 (bits 127:126).
    const v4u g0 = {0x80000001u, lds_base, (unsigned)(ga & 0xFFFFFFFFu),
                    (unsigned)(((ga >> 32) & 0x1FFFFFFu) | (2u << 30))};
    // D# group 1: data_size=4B (2<<16); tensor_dim0=128 (bits79:48);
    //             tensor_dim1=n (bits111:80); tile_dim0=128 (bits127:112);
    //             tile_dim1=m (#valid indices); tensor_dim0_stride=128.
    const v8i g1 = {0x00020000, (int)(128u << 16),
                    (int)(((unsigned)n & 0xFFFFu) << 16),
                    (int)((((unsigned)n >> 16) & 0xFFFFu) | (128u << 16)),
                    m, 128, 0, 0};
    const v4i g2 = {(int)idx[0], (int)idx[1], (int)idx[2], (int)idx[3]};
    const v4i g3 = {(int)idx[4], (int)idx[5], (int)idx[6], (int)idx[7]};
    const v8i gz = {0, 0, 0, 0, 0, 0, 0, 0};  // unused 5th group slot (VADDR4)

    // clang-23 / therock lane: 6-arg form (v4u, v8i, v4i, v4i, v8i, i32 cpol).
    __builtin_amdgcn_tensor_load_to_lds(g0, g1, g2, g3, gz, 0);  // TENSORcnt++
    __builtin_amdgcn_s_wait_tensorcnt(0);                        // rows in LDS
    asm volatile("" ::: "memory");  // pin ds reads after the tensor wait

    for (int i = 0; i < m; ++i) {
      const int s = __builtin_amdgcn_readlane(sl, i);
      const int dso = deg_out[s];  // scalar load (s is SGPR-uniform)
      const float ro = rsqrtf((float)(dso > 1 ? dso : 1));
      const float4 hs = ((const float4*)(stage + i * 128))[lane];  // ds_load_b128
      acc.x = fmaf(hs.x, ro, acc.x);
      acc.y = fmaf(hs.y, ro, acc.y);
      acc.z = fmaf(hs.z, ro, acc.z);
      acc.w = fmaf(hs.w, ro, acc.w);
    }
#else
    // Fallback: direct global gather with per-lane prefetch of the next row.
    for (int i = 0; i < m; ++i) {
      const int s = __builtin_amdgcn_readlane(sl, i);
      if (i + 1 < m) {
        const int sn = __builtin_amdgcn_readlane(sl, i + 1);
        __builtin_prefetch((const char*)(H + (size_t)sn * 128) + lane * 16, 0, 3);
      }
      const int dso = deg_out[s];
      const float ro = rsqrtf((float)(dso > 1 ? dso : 1));
      const float4 hs = ((const float4*)(H + (size_t)s * 128))[lane];
      acc.x = fmaf(hs.x, ro, acc.x);
      acc.y = fmaf(hs.y, ro, acc.y);
      acc.z = fmaf(hs.z, ro, acc.z);
      acc.w = fmaf(hs.w, ro, acc.w);
    }
#endif
  }

  const float rin = rsqrtf((float)(cnt > 1 ? cnt : 1));
  float4 o;
  o.x = h.x - kc.x * (h.x - acc.x * rin);
  o.y = h.y - kc.y * (h.y - acc.y * rin);
  o.z = h.z - kc.z * (h.z - acc.z * rin);
  o.w = h.w - kc.w * (h.w - acc.w * rin);
  ((float4*)(out + (size_t)node * 128))[lane] = o;
}

extern "C" void kernel_launch(void* const* d_in, const int* in_sizes, int n_in,
                              void* d_out, int out_size, void* d_ws, size_t ws_size,
                              hipStream_t stream) {
  const float* H  = (const float*)d_in[0];  // [n,128] f32
  const float* K  = (const float*)d_in[1];  // [1,128] f32
  const int*  src = (const int*)d_in[2];    // [e]
  const int*  dst = (const int*)d_in[3];    // [e]
  float* out = (float*)d_out;

  const int n = in_sizes[0] / 128;
  const int e = in_sizes[2];

  // Workspace (ints): deg_out[n] | deg_in[n] | cursor[n] | offsets[n+1] | csr_src[e]
  int* w       = (int*)d_ws;
  int* deg_out = w;
  int* deg_in  = w + n;
  int* cursor  = w + 2 * n;
  int* offsets = w + 3 * n;
  int* csr     = w + 4 * n + 1;

  const int eb = (e + 255) / 256;
  const int nb = (n + WPB - 1) / WPB;

  k_zero  <<<512, 256, 0, stream>>>(w, 3 * n);
  k_deg   <<<eb, 256, 0, stream>>>(src, dst, deg_out, deg_in, e);
  k_scan  <<<1, 1024, 0, stream>>>(deg_in, offsets, n);
  k_bucket<<<eb, 256, 0, stream>>>(src, dst, offsets, cursor, csr, e);
  k_agg   <<<nb, WPB * 32, 0, stream>>>(H, K, offsets, deg_out, csr, out, n);
}